// Musicmamba_154618823210
// MI455X (gfx1250) — compile-verified
//
#include <hip/hip_runtime.h>
#include <hip/hip_bf16.h>
#include <math.h>

// ---------------- model constants (match reference) ----------------
static constexpr int K_ = 4;        // token streams / heads
static constexpr int S_ = 1024;     // sequence length
static constexpr int D_ = 1024;     // hidden
static constexpr int DI_ = 2048;    // intermediate
static constexpr int NST_ = 16;     // ssm state
static constexpr int DT_RANK_ = 64;
static constexpr int CONV_K_ = 4;
static constexpr int N_LAYERS_ = 48;
static constexpr int CARD_ = 2048;
static constexpr float EPS_ = 1e-5f;

typedef __attribute__((ext_vector_type(2))) float v2f;
typedef __attribute__((ext_vector_type(8))) float v8f;
typedef __attribute__((ext_vector_type(4))) unsigned int v4u;
typedef __attribute__((ext_vector_type(8))) int v8i;
typedef __attribute__((ext_vector_type(4))) int v4i;

// ---------------- embedding: x[s][d] = sum_k emb[k, seq[k][s], d] ----------------
__global__ void embed_kernel(float* __restrict__ X,
                             const int* __restrict__ seq,
                             const float* __restrict__ emb) {
    int idx = blockIdx.x * blockDim.x + threadIdx.x;   // over S_*D_
    if (idx >= S_ * D_) return;
    int s = idx >> 10;
    int d = idx & (D_ - 1);
    float acc = 0.f;
#pragma unroll
    for (int k = 0; k < K_; ++k) {
        int tok = seq[k * S_ + s];
        acc += emb[((size_t)k * CARD_ + tok) * D_ + d];
    }
    X[idx] = acc;
}

// ---------------- RMSNorm over last dim (D_=1024), one block per row ----------------
__global__ void rmsnorm_kernel(float* __restrict__ out,
                               const float* __restrict__ x,
                               const float* __restrict__ w) {
    __shared__ float red[256];
    int row = blockIdx.x;
    int tid = threadIdx.x;
    const float* xr = x + (size_t)row * D_;
    float s = 0.f;
    for (int i = tid; i < D_; i += 256) { float v = xr[i]; s += v * v; }
    red[tid] = s;
    __syncthreads();
    for (int off = 128; off > 0; off >>= 1) {
        if (tid < off) red[tid] += red[tid + off];
        __syncthreads();
    }
    float scale = rsqrtf(red[0] / (float)D_ + EPS_);
    float* orow = out + (size_t)row * D_;
    for (int i = tid; i < D_; i += 256) orow[i] = xr[i] * scale * w[i];
}

// ============================================================================
// TDM helper: DMA a (tile_rows x 16) fp32 tile (row stride ld_elems) from
// global to LDS with row padding (16 DW interval, 2 DW pad -> 18-float rows).
// D# packing per CDNA5 ISA 8.3/8.4. Rows past rows_rem are zero-filled (OOB).
// clang-23 toolchain: 6-arg builtin (g0, g1, g2, g3, extra, cpol).
// ============================================================================
static constexpr int KB_ = 16;   // K-slab (tile_dim0, floats)
static constexpr int RS_ = 18;   // padded LDS row stride (floats)

__device__ __forceinline__ void tdm_load_tile(unsigned lds_off, const float* gptr,
                                              unsigned rows_rem, unsigned cols_rem,
                                              unsigned ld_elems, unsigned tile_rows) {
    unsigned long long ga = (unsigned long long)(uintptr_t)gptr;
    v4u g0;
    g0.x = 1u;                                               // count=1 (valid), user mode
    g0.y = lds_off;                                          // lds_addr
    g0.z = (unsigned)(ga & 0xFFFFFFFFull);                   // global_addr[31:0]
    g0.w = (unsigned)((ga >> 32) & 0x01FFFFFFull) | (2u << 30); // addr[56:32] | type=2
    v8i g1;
    // data_size=4B(2), pad_enable, pad_interval=16DW(3), pad_amount=2DW(1)
    g1[0] = (int)((2u << 16) | (1u << 20) | (3u << 22) | (1u << 25));
    g1[1] = (int)((cols_rem & 0xFFFFu) << 16);                               // tensor_dim0 lo16
    g1[2] = (int)(((cols_rem >> 16) & 0xFFFFu) | ((rows_rem & 0xFFFFu) << 16)); // dim0 hi | dim1 lo
    g1[3] = (int)(((rows_rem >> 16) & 0xFFFFu) | ((unsigned)KB_ << 16));     // dim1 hi | tile_dim0
    g1[4] = (int)(tile_rows & 0xFFFFu);                                      // tile_dim1 (tile_dim2=0)
    g1[5] = (int)ld_elems;                                                   // tensor_dim0_stride lo32
    g1[6] = 0;
    g1[7] = 0;
    v4i z4 = {0, 0, 0, 0};
    v8i z8 = {0, 0, 0, 0, 0, 0, 0, 0};
    __builtin_amdgcn_tensor_load_to_lds(g0, g1, z4, z4, z8, 0);
}

// ============================================================================
// TDM-staged fp32 WMMA GEMM: out[M,N] = act[M,K] @ W[N,K]^T (+bias/softplus/residual)
// Block (8 waves) -> 128x128 tile; wave -> 64(M) x 32(N); double-buffered LDS
// slabs of K=16 loaded by TENSOR_LOAD_TO_LDS (wave 0), waited on TENSORcnt.
// V_WMMA_F32_16X16X4_F32 fragments (ISA 7.12.2):
//   A/B: lane L holds row (L&15), elems k + 2*(L>>4) + {0,1}  -> 8B LDS load
//   C/D: vgpr r, lane L -> row r + 8*(L>>4), col L&15
// ============================================================================
__global__ void __launch_bounds__(256)
gemm_tdm_kernel(float* __restrict__ out, int ldo,
                const float* __restrict__ act, int lda,
                const float* __restrict__ W,
                const float* __restrict__ bias,
                const float* __restrict__ residual,
                int M, int N, int Kd, int do_softplus) {
    __shared__ float As[2][128 * RS_];
    __shared__ float Ws[2][128 * RS_];

    int wid  = threadIdx.x >> 5;
    int lane = threadIdx.x & 31;
    int l16  = lane & 15;
    int hs   = lane >> 4;
    int mh   = wid >> 2;    // 0..1 : row half
    int nq   = wid & 3;     // 0..3 : col quarter
    int m0   = blockIdx.y * 128;
    int n0   = blockIdx.x * 128;

    v8f acc[4][2];
    v8f zero = {0.f, 0.f, 0.f, 0.f, 0.f, 0.f, 0.f, 0.f};
#pragma unroll
    for (int t = 0; t < 4; ++t) { acc[t][0] = zero; acc[t][1] = zero; }

    unsigned rowsA = (unsigned)(M - m0);
    unsigned rowsW = (unsigned)(N - n0);
    const float* aTile = act + (size_t)m0 * lda;
    const float* wTile = W + (size_t)n0 * Kd;

    if (wid == 0) {
        tdm_load_tile((unsigned)(uintptr_t)&As[0][0], aTile, rowsA, (unsigned)Kd,
                      (unsigned)lda, 128u);
        tdm_load_tile((unsigned)(uintptr_t)&Ws[0][0], wTile, rowsW, (unsigned)Kd,
                      (unsigned)Kd, 128u);
    }

    int cur = 0;
    for (int k0 = 0; k0 < Kd; k0 += KB_) {
        if (wid == 0) __builtin_amdgcn_s_wait_tensorcnt(0);
        __syncthreads();                      // slab `cur` ready for all waves
        if (wid == 0 && (k0 + KB_) < Kd) {    // prefetch next slab (async DMA)
            tdm_load_tile((unsigned)(uintptr_t)&As[cur ^ 1][0], aTile + k0 + KB_,
                          rowsA, (unsigned)(Kd - k0 - KB_), (unsigned)lda, 128u);
            tdm_load_tile((unsigned)(uintptr_t)&Ws[cur ^ 1][0], wTile + k0 + KB_,
                          rowsW, (unsigned)(Kd - k0 - KB_), (unsigned)Kd, 128u);
        }
        const float* ap = &As[cur][(mh * 64 + l16) * RS_ + 2 * hs];
        const float* wp = &Ws[cur][(nq * 32 + l16) * RS_ + 2 * hs];
#pragma unroll
        for (int kk = 0; kk < KB_; kk += 4) {
            v2f b0 = *(const v2f*)(wp + kk);
            v2f b1 = *(const v2f*)(wp + 16 * RS_ + kk);
#pragma unroll
            for (int t = 0; t < 4; ++t) {
                v2f a = *(const v2f*)(ap + t * 16 * RS_ + kk);
                acc[t][0] = __builtin_amdgcn_wmma_f32_16x16x4_f32(
                    false, a, false, b0, (short)0, acc[t][0], false, false);
                acc[t][1] = __builtin_amdgcn_wmma_f32_16x16x4_f32(
                    false, a, false, b1, (short)0, acc[t][1], false, false);
            }
        }
        cur ^= 1;
    }

#pragma unroll
    for (int j = 0; j < 2; ++j) {
        int col = n0 + nq * 32 + j * 16 + l16;
        if (col < N) {
            float bv = bias ? bias[col] : 0.f;
#pragma unroll
            for (int t = 0; t < 4; ++t) {
#pragma unroll
                for (int r = 0; r < 8; ++r) {
                    int row = m0 + mh * 64 + t * 16 + hs * 8 + r;
                    float v = acc[t][j][r] + bv;
                    if (do_softplus) v = (v > 20.f) ? v : log1pf(expf(v));
                    if (residual) v += residual[(size_t)row * ldo + col];
                    out[(size_t)row * ldo + col] = v;
                }
            }
        }
    }
}

// ---------------- direct-load WMMA GEMM (for the tiny N=96 x-proj) ----------------
__global__ void gemm_wmma_kernel(float* __restrict__ out, int ldo,
                                 const float* __restrict__ act, int lda,
                                 const float* __restrict__ W,
                                 int M, int N, int Kd) {
    int wid  = threadIdx.x >> 5;
    int lane = threadIdx.x & 31;
    int gw   = blockIdx.x * 8 + wid;
    int ntiles  = N >> 4;
    int mgroups = M >> 6;
    if (gw >= ntiles * mgroups) return;
    int n0 = (gw % ntiles) << 4;
    int m0 = (gw / ntiles) << 6;
    int l16 = lane & 15;
    int hs  = lane >> 4;

    v8f acc[4];
    v8f zero = {0.f, 0.f, 0.f, 0.f, 0.f, 0.f, 0.f, 0.f};
#pragma unroll
    for (int t = 0; t < 4; ++t) acc[t] = zero;

    const float* wrow = W + (size_t)(n0 + l16) * Kd + 2 * hs;
    const float* arow = act + (size_t)(m0 + l16) * lda + 2 * hs;
    size_t astep = (size_t)16 * lda;

    for (int k = 0; k < Kd; k += 4) {
        v2f b = *(const v2f*)(wrow + k);
#pragma unroll
        for (int t = 0; t < 4; ++t) {
            v2f a = *(const v2f*)(arow + (size_t)t * astep + k);
            acc[t] = __builtin_amdgcn_wmma_f32_16x16x4_f32(
                false, a, false, b, (short)0, acc[t], false, false);
        }
    }
    int col = n0 + l16;
#pragma unroll
    for (int t = 0; t < 4; ++t) {
#pragma unroll
        for (int r = 0; r < 8; ++r) {
            int row = m0 + t * 16 + hs * 8 + r;
            out[(size_t)row * ldo + col] = acc[t][r];
        }
    }
}

// ---------------- causal depthwise conv (k=4) + bias + silu ----------------
__global__ void conv_silu_kernel(float* __restrict__ u,
                                 const float* __restrict__ proj,
                                 const float* __restrict__ cw,
                                 const float* __restrict__ cb) {
    int idx = blockIdx.x * blockDim.x + threadIdx.x;
    if (idx >= S_ * DI_) return;
    int s = idx / DI_;
    int d = idx - s * DI_;
    float acc = cb[d];
#pragma unroll
    for (int j = 0; j < CONV_K_; ++j) {
        int sp = s - (CONV_K_ - 1) + j;
        if (sp >= 0) acc += cw[d * CONV_K_ + j] * proj[(size_t)sp * (2 * DI_) + d];
    }
    u[idx] = acc / (1.f + expf(-acc));
}

// ---------------- selective scan: sequential over S, parallel over DI channels --------
__global__ void scan_kernel(float* __restrict__ y,
                            const float* __restrict__ dt,
                            const float* __restrict__ ssm,    // (S,96): [64:80]=B, [80:96]=C
                            const float* __restrict__ u,
                            const float* __restrict__ proj,   // gate = proj[:, DI_:]
                            const float* __restrict__ A_log,
                            const float* __restrict__ D_skip) {
    int d = blockIdx.x * blockDim.x + threadIdx.x;
    float A[NST_], h[NST_];
#pragma unroll
    for (int n = 0; n < NST_; ++n) {
        A[n] = -expf(A_log[(size_t)d * NST_ + n]);
        h[n] = 0.f;
    }
    float dsk = D_skip[d];
    __shared__ float sB[NST_];
    __shared__ float sC[NST_];
    for (int s = 0; s < S_; ++s) {
        __syncthreads();
        if (threadIdx.x < NST_)
            sB[threadIdx.x] = ssm[(size_t)s * 96 + DT_RANK_ + threadIdx.x];
        else if (threadIdx.x < 2 * NST_)
            sC[threadIdx.x - NST_] = ssm[(size_t)s * 96 + DT_RANK_ + NST_ + (threadIdx.x - NST_)];
        __syncthreads();
        float dtv = dt[(size_t)s * DI_ + d];
        float uv  = u[(size_t)s * DI_ + d];
        float du  = dtv * uv;
        float acc = 0.f;
#pragma unroll
        for (int n = 0; n < NST_; ++n) {
            h[n] = expf(dtv * A[n]) * h[n] + du * sB[n];
            acc += h[n] * sC[n];
        }
        float g  = proj[(size_t)s * (2 * DI_) + DI_ + d];
        float sg = g / (1.f + expf(-g));
        y[(size_t)s * DI_ + d] = (acc + uv * dsk) * sg;
    }
}

// ---------------- host-side launchers ----------------
static inline void launch_gemm_tdm(float* out, int ldo, const float* act, int lda,
                                   const float* W, const float* bias, const float* residual,
                                   int M, int N, int Kd, int softplus, hipStream_t stream) {
    dim3 grid((N + 127) / 128, M / 128);
    gemm_tdm_kernel<<<grid, 256, 0, stream>>>(out, ldo, act, lda, W, bias, residual,
                                              M, N, Kd, softplus);
}

extern "C" void kernel_launch(void* const* d_in, const int* in_sizes, int n_in,
                              void* d_out, int out_size, void* d_ws, size_t ws_size,
                              hipStream_t stream) {
    const int*   seq     = (const int*)  d_in[0];
    const float* emb_W   = (const float*)d_in[1];
    const float* norm_w  = (const float*)d_in[2];
    const float* in_W    = (const float*)d_in[3];
    const float* conv_w  = (const float*)d_in[4];
    const float* conv_b  = (const float*)d_in[5];
    const float* x_W     = (const float*)d_in[6];
    const float* dt_W    = (const float*)d_in[7];
    const float* dt_b    = (const float*)d_in[8];
    const float* A_log   = (const float*)d_in[9];
    const float* D_skip  = (const float*)d_in[10];
    const float* out_W   = (const float*)d_in[11];
    const float* normf_w = (const float*)d_in[12];
    const float* head_W  = (const float*)d_in[13];
    const float* head_b  = (const float*)d_in[14];
    float* out = (float*)d_out;

    // workspace carve-up (~50 MB)
    char* w = (char*)d_ws;
    float* X    = (float*)(w + 0);              // (S, D)    4 MB
    float* H    = (float*)(w + (4u  << 20));    // (S, D)    4 MB
    float* PROJ = (float*)(w + (8u  << 20));    // (S, 2*DI) 16 MB
    float* U    = (float*)(w + (24u << 20));    // (S, DI)   8 MB
    float* SSM  = (float*)(w + (32u << 20));    // (S, 96)   0.4 MB
    float* DT   = (float*)(w + (33u << 20));    // (S, DI)   8 MB
    float* Y    = (float*)(w + (41u << 20));    // (S, DI)   8 MB

    embed_kernel<<<(S_ * D_ + 255) / 256, 256, 0, stream>>>(X, seq, emb_W);

    for (int l = 0; l < N_LAYERS_; ++l) {
        const float* nw  = norm_w + (size_t)l * D_;
        const float* iW  = in_W   + (size_t)l * (2 * DI_) * D_;
        const float* cw  = conv_w + (size_t)l * DI_ * CONV_K_;
        const float* cb  = conv_b + (size_t)l * DI_;
        const float* xW  = x_W    + (size_t)l * (DT_RANK_ + 2 * NST_) * DI_;
        const float* dW  = dt_W   + (size_t)l * DI_ * DT_RANK_;
        const float* db  = dt_b   + (size_t)l * DI_;
        const float* al  = A_log  + (size_t)l * DI_ * NST_;
        const float* dsk = D_skip + (size_t)l * DI_;
        const float* oW  = out_W  + (size_t)l * D_ * DI_;

        rmsnorm_kernel<<<S_, 256, 0, stream>>>(H, X, nw);
        // proj = h @ in_W^T   (1024 x 4096 x 1024) -- TDM-staged WMMA
        launch_gemm_tdm(PROJ, 2 * DI_, H, D_, iW, nullptr, nullptr, S_, 2 * DI_, D_, 0, stream);
        conv_silu_kernel<<<(S_ * DI_ + 255) / 256, 256, 0, stream>>>(U, PROJ, cw, cb);
        // ssm_in = u @ x_W^T  (1024 x 96 x 2048) -- small N: direct WMMA
        {
            int waves = ((DT_RANK_ + 2 * NST_) >> 4) * (S_ >> 6);
            gemm_wmma_kernel<<<(waves + 7) / 8, 256, 0, stream>>>(
                SSM, DT_RANK_ + 2 * NST_, U, DI_, xW, S_, DT_RANK_ + 2 * NST_, DI_);
        }
        // dt = softplus(dt_r @ dt_W^T + dt_b)  (1024 x 2048 x 64), act stride 96
        launch_gemm_tdm(DT, DI_, SSM, DT_RANK_ + 2 * NST_, dW, db, nullptr,
                        S_, DI_, DT_RANK_, 1, stream);
        scan_kernel<<<DI_ / 256, 256, 0, stream>>>(Y, DT, SSM, U, PROJ, al, dsk);
        // x = x + y @ out_W^T  (1024 x 1024 x 2048)
        launch_gemm_tdm(X, D_, Y, DI_, oW, nullptr, X, S_, D_, DI_, 0, stream);
    }

    rmsnorm_kernel<<<S_, 256, 0, stream>>>(H, X, normf_w);

    for (int k = 0; k < K_; ++k) {
        launch_gemm_tdm(out + (size_t)k * S_ * CARD_, CARD_, H, D_,
                        head_W + (size_t)k * CARD_ * D_, head_b + (size_t)k * CARD_,
                        nullptr, S_, CARD_, D_, 0, stream);
    }
    (void)in_sizes; (void)n_in; (void)out_size; (void)ws_size;
}